// IntraLoss_29111288332478
// MI455X (gfx1250) — compile-verified
//
#include <hip/hip_runtime.h>
#include <math.h>

// CDNA5 / gfx1250: wave32, WMMA (not MFMA).
typedef __attribute__((ext_vector_type(2))) float v2f;
typedef __attribute__((ext_vector_type(8))) float v8f;

#define TILE_M          16
#define WAVES_PER_BLOCK 8
#define THREADS         (WAVES_PER_BLOCK * 32)
#define DIM             512

// Kernel 1: each wave owns a 16-row tile. dist2(row) = diagonal of Diff*Diff^T,
// accumulated with v_wmma_f32_16x16x4_f32 (A and B fed the same registers:
// the f32 A-layout and B-layout are index-transposes, so B==A means B=A^T).
__global__ __launch_bounds__(THREADS) void intra_loss_tile_kernel(
    const float* __restrict__ features,
    const int*   __restrict__ labels,
    const float* __restrict__ center,
    float*       __restrict__ block_partials)
{
    const int tid  = threadIdx.x;
    const int lane = tid & 31;
    const int wave = tid >> 5;
    const int tile = blockIdx.x * WAVES_PER_BLOCK + wave;

    const int r     = lane & 15;           // row within 16-row tile
    const int khalf = (lane >> 4) << 1;    // lanes 0-15 -> K+{0,1}, lanes 16-31 -> K+{2,3}
    const int row   = tile * TILE_M + r;

    const int lab = labels[row];
    const float* __restrict__ f = features + (size_t)row * DIM + khalf;
    const float* __restrict__ c = center   + (size_t)lab * DIM + khalf;

    v8f acc = {};  // 16x16 f32 Gram accumulator (only diagonal is consumed)

    for (int kk = 0; kk < DIM; kk += 32) {
        // global_prefetch_b8 one outer iteration ahead
        __builtin_prefetch(f + kk + 32, 0, 0);
        __builtin_prefetch(c + kk + 32, 0, 0);
#pragma unroll
        for (int k = kk; k < kk + 32; k += 4) {
            const v2f fv = *(const v2f*)(f + k);
            const v2f cv = *(const v2f*)(c + k);
            const v2f a  = fv - cv;   // diff chunk in A-layout (== B-layout of diff^T)
            // 8-arg form: (neg_a, A, neg_b, B, c_mod, C, reuse_a, reuse_b)
            acc = __builtin_amdgcn_wmma_f32_16x16x4_f32(
                false, a, false, a, (short)0, acc, false, false);
        }
    }

    // Diagonal extraction from C/D layout:
    //  (m,m) for m=0..7  -> VGPR m,   lane m
    //  (m,m) for m=8..15 -> VGPR m-8, lane m+16 (N = lane-16 = m => lane = m+16..)
    float d2 = 0.0f;
#pragma unroll
    for (int v = 0; v < 8; ++v) {
        const bool sel = (lane == v) || (lane == v + 24);
        d2 = sel ? acc[v] : d2;
    }
    const bool active = (lane < 8) || (lane >= 24);   // 16 lanes own the 16 rows
    const float part  = active ? sqrtf(fmaxf(d2, 0.0f)) : 0.0f;

    // Deterministic in-block tree reduction.
    __shared__ float sdata[THREADS];
    sdata[tid] = part;
    __syncthreads();
#pragma unroll
    for (int s = THREADS / 2; s > 0; s >>= 1) {
        if (tid < s) sdata[tid] += sdata[tid + s];
        __syncthreads();
    }
    if (tid == 0) block_partials[blockIdx.x] = sdata[0];
}

// Kernel 2: single-block deterministic reduction of per-block partials.
__global__ __launch_bounds__(256) void final_reduce_kernel(
    const float* __restrict__ partials, int nblocks, float inv_n,
    float* __restrict__ out)
{
    __shared__ float sdata[256];
    const int tid = threadIdx.x;
    float s = 0.0f;
    for (int i = tid; i < nblocks; i += 256) s += partials[i];
    sdata[tid] = s;
    __syncthreads();
#pragma unroll
    for (int st = 128; st > 0; st >>= 1) {
        if (tid < st) sdata[tid] += sdata[tid + st];
        __syncthreads();
    }
    if (tid == 0) out[0] = sdata[0] * inv_n;
}

extern "C" void kernel_launch(void* const* d_in, const int* in_sizes, int n_in,
                              void* d_out, int out_size, void* d_ws, size_t ws_size,
                              hipStream_t stream) {
    const float* features = (const float*)d_in[0];   // [N, 512] f32
    const int*   labels   = (const int*)d_in[1];     // [N] int
    const float* center   = (const float*)d_in[2];   // [1000, 512] f32
    float* out = (float*)d_out;
    float* ws  = (float*)d_ws;

    const int n      = in_sizes[1];                  // N = 32768
    const int tiles  = n / TILE_M;                   // 2048
    int blocks       = tiles / WAVES_PER_BLOCK;      // 256
    if (blocks < 1) blocks = 1;

    intra_loss_tile_kernel<<<blocks, THREADS, 0, stream>>>(features, labels, center, ws);
    final_reduce_kernel<<<1, 256, 0, stream>>>(ws, blocks, 1.0f / (float)n, out);
}